// ChessTacticGNN_13469017440463
// MI455X (gfx1250) — compile-verified
//
#include <hip/hip_runtime.h>
#include <math.h>

// GCN on MI455X (gfx1250):
//  - fp32 WMMA (v_wmma_f32_16x16x4_f32) for the feature GEMMs (exact fp32 math)
//  - native global_atomic_add_f32 for the edge scatter (L2-resident: 32MB
//    feature matrices << 192MB L2, so random traffic stays on-die)
//  - wave32 tiling: one wave -> 16x64 output tile, 4 x v8f accumulators

typedef float v2f __attribute__((ext_vector_type(2)));
typedef float v8f __attribute__((ext_vector_type(8)));

#define HIDC 64

__device__ __forceinline__ void atomAddF(float* p, float v) {
  unsafeAtomicAdd(p, v);   // lowers to global_atomic_add_f32 on gfx1250
}

// ---------------------------------------------------------------- utilities
__global__ void k_zero(float* __restrict__ p, int n) {
  int i = blockIdx.x * blockDim.x + threadIdx.x;
  if (i < n) p[i] = 0.0f;
}

// deg[dst] += 1 per edge (deg buffer pre-zeroed)
__global__ void k_degree(const int* __restrict__ dst, float* __restrict__ deg, int E) {
  int e = blockIdx.x * blockDim.x + threadIdx.x;
  if (e < E) atomAddF(&deg[dst[e]], 1.0f);
}

// in-place: dinv = rsqrt(deg + 1)   (self loop adds 1 to degree)
__global__ void k_dinv(float* __restrict__ d, int n) {
  int i = blockIdx.x * blockDim.x + threadIdx.x;
  if (i < n) d[i] = rsqrtf(d[i] + 1.0f);
}

// norm[e] = dinv[src]*dinv[dst]  (reused by both layers)
__global__ void k_norm(const int* __restrict__ src, const int* __restrict__ dst,
                       const float* __restrict__ dinv, float* __restrict__ norm, int E) {
  int e = blockIdx.x * blockDim.x + threadIdx.x;
  if (e < E) norm[e] = dinv[src[e]] * dinv[dst[e]];
}

// ------------------------------------------------- WMMA GEMM: H[M,64]=X[M,K]@W[K,64]
// One wave computes a 16x64 tile. A,B fragment layouts per CDNA5 ISA 7.12.2:
//   A 16x4 f32: lanes 0-15 -> M rows, K={0,1}; lanes 16-31 -> same rows, K={2,3}
//   B 4x16 f32: symmetric (K rows striped the same way, N across lanes)
//   C/D 16x16 f32: vgpr i holds M = i + 8*(lane>=16), N = lane&15
template <int K>
__global__ void k_gemm_wmma(const float* __restrict__ X, const float* __restrict__ W,
                            float* __restrict__ H, int M) {
  const int lane  = threadIdx.x & 31;
  const int wave  = threadIdx.x >> 5;
  const int mtile = blockIdx.x * (blockDim.x >> 5) + wave;
  const int m0    = mtile << 4;
  if (m0 >= M) return;                 // wave-uniform: EXEC stays all-1s for WMMA

  const int l15   = lane & 15;
  const int khalf = (lane >> 4) << 1;  // 0 or 2
  const float* xrow = X + (size_t)(m0 + l15) * K;

  v8f acc0 = {}, acc1 = {}, acc2 = {}, acc3 = {};
#pragma unroll
  for (int k0 = 0; k0 < K; k0 += 4) {
    v2f a;
    a.x = xrow[k0 + khalf];
    a.y = xrow[k0 + khalf + 1];
    const float* w0 = W + (size_t)(k0 + khalf) * HIDC + l15;
    const float* w1 = w0 + HIDC;
    v2f b;
    b.x = w0[0];  b.y = w1[0];
    acc0 = __builtin_amdgcn_wmma_f32_16x16x4_f32(false, a, false, b, (short)0, acc0, false, false);
    b.x = w0[16]; b.y = w1[16];
    acc1 = __builtin_amdgcn_wmma_f32_16x16x4_f32(false, a, false, b, (short)0, acc1, false, false);
    b.x = w0[32]; b.y = w1[32];
    acc2 = __builtin_amdgcn_wmma_f32_16x16x4_f32(false, a, false, b, (short)0, acc2, false, false);
    b.x = w0[48]; b.y = w1[48];
    acc3 = __builtin_amdgcn_wmma_f32_16x16x4_f32(false, a, false, b, (short)0, acc3, false, false);
  }

  float* hout = H + (size_t)m0 * HIDC;
  const int mb = (lane >> 4) << 3;     // 0 or 8
#pragma unroll
  for (int i = 0; i < 8; ++i) {
    float* row = hout + (size_t)(mb + i) * HIDC + l15;
    row[0]  = acc0[i];
    row[16] = acc1[i];
    row[32] = acc2[i];
    row[48] = acc3[i];
  }
}

// -------------------------------------------- edge scatter: agg[dst] += h[src]*norm
// thread = (edge, channel); total = E*64
__global__ void k_scatter(const int* __restrict__ src, const int* __restrict__ dst,
                          const float* __restrict__ norm, const float* __restrict__ h,
                          float* __restrict__ agg, int total) {
  int tid = blockIdx.x * blockDim.x + threadIdx.x;
  if (tid >= total) return;
  int e = tid >> 6, c = tid & 63;
  int s = src[e], d = dst[e];
  atomAddF(&agg[(size_t)d * HIDC + c], h[(size_t)s * HIDC + c] * norm[e]);
}

// h = relu(agg + h*dinv^2 + b)   (in-place, elementwise)
__global__ void k_combine_relu(const float* __restrict__ agg, const float* __restrict__ dinv,
                               const float* __restrict__ b, float* __restrict__ h, int total) {
  int tid = blockIdx.x * blockDim.x + threadIdx.x;
  if (tid >= total) return;
  int i = tid >> 6, c = tid & 63;
  float di = dinv[i];
  float v  = agg[tid] + h[tid] * di * di + b[c];
  h[tid]   = v > 0.0f ? v : 0.0f;
}

// pool sums/counts (sums,cnt pre-zeroed)
__global__ void k_pool(const float* __restrict__ h, const int* __restrict__ batch,
                       float* __restrict__ sums, float* __restrict__ cnt, int total) {
  int tid = blockIdx.x * blockDim.x + threadIdx.x;
  if (tid >= total) return;
  int i = tid >> 6, c = tid & 63;
  int g = batch[i];
  atomAddF(&sums[(size_t)g * HIDC + c], h[tid]);
  if (c == 0) atomAddF(&cnt[g], 1.0f);
}

// out[g,o] = sigmoid( (sums[g,:]/max(cnt,1)) . Wfc[:,o] + bfc[o] )
__global__ void k_head(const float* __restrict__ sums, const float* __restrict__ cnt,
                       const float* __restrict__ Wfc, const float* __restrict__ bfc,
                       float* __restrict__ out, int G, int OUTC) {
  int tid = blockIdx.x * blockDim.x + threadIdx.x;
  if (tid >= G * OUTC) return;
  int g = tid / OUTC, o = tid % OUTC;
  float inv = 1.0f / fmaxf(cnt[g], 1.0f);
  float v = bfc[o];
#pragma unroll 8
  for (int c = 0; c < HIDC; ++c)
    v += sums[(size_t)g * HIDC + c] * inv * Wfc[c * OUTC + o];
  out[tid] = 1.0f / (1.0f + expf(-v));
}

// ------------------------------------------------------------------ launcher
extern "C" void kernel_launch(void* const* d_in, const int* in_sizes, int n_in,
                              void* d_out, int out_size, void* d_ws, size_t ws_size,
                              hipStream_t stream) {
  const float* x   = (const float*)d_in[0];
  const int*   ei  = (const int*)  d_in[1];   // [2,E] flat: src then dst
  const int*   bat = (const int*)  d_in[2];
  const float* W1  = (const float*)d_in[3];
  const float* b1  = (const float*)d_in[4];
  const float* W2  = (const float*)d_in[5];
  const float* b2  = (const float*)d_in[6];
  const float* Wfc = (const float*)d_in[7];
  const float* bfc = (const float*)d_in[8];
  float* out = (float*)d_out;

  const int N    = in_sizes[2];        // 131072 nodes
  const int E    = in_sizes[1] / 2;    // 2097152 edges
  const int OUTC = in_sizes[8];        // 4
  const int G    = out_size / OUTC;    // 2048 graphs
  const int* src = ei;
  const int* dst = ei + E;

  char* ws = (char*)d_ws;
  float* dinv = (float*)ws;  ws += (size_t)N * 4;
  float* norm = (float*)ws;  ws += (size_t)E * 4;
  float* hA   = (float*)ws;  ws += (size_t)N * HIDC * 4;
  float* hB   = (float*)ws;  ws += (size_t)N * HIDC * 4;
  float* agg  = (float*)ws;  ws += (size_t)N * HIDC * 4;
  float* sums = (float*)ws;  ws += (size_t)G * HIDC * 4;
  float* cnt  = (float*)ws;  ws += (size_t)G * 4;

  const int T = 256;
  auto nb = [](int n, int t) { return (n + t - 1) / t; };
  const int NH = N * HIDC;             // 8.4M, fits int
  const int EH = E * HIDC;             // 134M, fits int
  const int gemmBlocks = (N / 16 + 7) / 8;   // 8 waves/block, 16 rows/wave

  // degrees -> dinv -> per-edge norm (shared by both layers)
  k_zero  <<<nb(N, T), T, 0, stream>>>(dinv, N);
  k_degree<<<nb(E, T), T, 0, stream>>>(dst, dinv, E);
  k_dinv  <<<nb(N, T), T, 0, stream>>>(dinv, N);
  k_norm  <<<nb(E, T), T, 0, stream>>>(src, dst, dinv, norm, E);

  // layer 1: hA = x@W1 ; agg = scatter ; hA = relu(agg + hA*dinv^2 + b1)
  k_gemm_wmma<12><<<gemmBlocks, 256, 0, stream>>>(x, W1, hA, N);
  k_zero        <<<nb(NH, T), T, 0, stream>>>(agg, NH);
  k_scatter     <<<nb(EH, T), T, 0, stream>>>(src, dst, norm, hA, agg, EH);
  k_combine_relu<<<nb(NH, T), T, 0, stream>>>(agg, dinv, b1, hA, NH);

  // layer 2
  k_gemm_wmma<64><<<gemmBlocks, 256, 0, stream>>>(hA, W2, hB, N);
  k_zero        <<<nb(NH, T), T, 0, stream>>>(agg, NH);
  k_scatter     <<<nb(EH, T), T, 0, stream>>>(src, dst, norm, hB, agg, EH);
  k_combine_relu<<<nb(NH, T), T, 0, stream>>>(agg, dinv, b2, hB, NH);

  // mean pool + linear head + sigmoid
  k_zero<<<nb(G * HIDC + G, T), T, 0, stream>>>(sums, G * HIDC + G); // sums+cnt contiguous
  k_pool<<<nb(NH, T), T, 0, stream>>>(hB, bat, sums, cnt, NH);
  k_head<<<nb(G * OUTC, T), T, 0, stream>>>(sums, cnt, Wfc, bfc, out, G, OUTC);
}